// DetectionBaseline_12953621364749
// MI455X (gfx1250) — compile-verified
//
#include <hip/hip_runtime.h>
#include <stdint.h>

// ---------------------------------------------------------------------------
// SSD detection post-processing for MI455X (gfx1250, wave32).
//  K1 decode+softmax  -> boxes[B,A,4], area[B,A], cls[B,Cf,A]
//  K2 overlap bits    -> ov[B,A,40]  (bit j of row i: iou(i,j) > 0.45)
//                        union term area_i+area_j via V_WMMA_F32_16X16X4_F32
//  K3 per-(b,c) sort  -> order[B,Cf,A]   (bitonic, 2048 pad, LDS)
//  K4 NMS scan        -> masked[B,Cf*A]  (1 wave/chain, bpermute broadcast,
//                        global_prefetch_b8 one 32-step chunk ahead)
//  K5 top-k           -> d_out           (bitonic over 32768 keys, 256KB LDS)
// ---------------------------------------------------------------------------

typedef float v2f __attribute__((ext_vector_type(2)));
typedef float v8f __attribute__((ext_vector_type(8)));

#define AN 1280        // anchors
#define NCLS 21
#define CF 20
#define ROWW 40        // u32 words per overlap row (1280 bits)

// ---------------- K1: decode boxes + softmax scores ----------------
__global__ void k_decode(const float* __restrict__ locs,
                         const float* __restrict__ scores,
                         const float* __restrict__ anchors,
                         float* __restrict__ boxes,
                         float* __restrict__ area,
                         float* __restrict__ cls,
                         int B) {
  int g = blockIdx.x * blockDim.x + threadIdx.x;
  if (g >= B * AN) return;
  int b = g / AN, a = g - b * AN;

  const float* l  = locs + (size_t)g * 4;
  const float* an = anchors + (size_t)a * 4;
  float acx = an[0], acy = an[1], aw = an[2], ah = an[3];
  float cx = l[0] * aw * 0.1f + acx;
  float cy = l[1] * ah * 0.1f + acy;
  float w  = expf(l[2] * 0.2f) * aw;
  float h  = expf(l[3] * 0.2f) * ah;
  float x1 = cx - 0.5f * w, y1 = cy - 0.5f * h;
  float x2 = cx + 0.5f * w, y2 = cy + 0.5f * h;
  float* bp = boxes + (size_t)g * 4;
  bp[0] = x1; bp[1] = y1; bp[2] = x2; bp[3] = y2;
  area[g] = (x2 - x1) * (y2 - y1);

  // softmax over 21 classes
  const float* s = scores + (size_t)g * NCLS;
  float mx = s[0];
  #pragma unroll
  for (int c = 1; c < NCLS; ++c) mx = fmaxf(mx, s[c]);
  float e[NCLS], sum = 0.f;
  #pragma unroll
  for (int c = 0; c < NCLS; ++c) { e[c] = expf(s[c] - mx); sum += e[c]; }
  float inv = 1.0f / sum;
  #pragma unroll
  for (int c = 1; c < NCLS; ++c)
    cls[((size_t)b * CF + (c - 1)) * AN + a] = e[c] * inv;
}

// ---------------- K2: overlap bit-matrix via WMMA ----------------
// One wave per 16x16 IoU tile.  union = (area_i + area_j) - inter; the outer
// sum is a rank-2 matmul on the matrix unit:
//   A(16x4) = [area_i, 1, 0, 0]   B(4x16) = [1...1 ; area_j ; 0 ; 0]
// All operands that depend on the (undocumented) half-wave striping of the
// B fragment are zero, so the result is layout-robust.
__global__ void k_overlap(const float* __restrict__ boxes,
                          const float* __restrict__ area,
                          uint32_t* __restrict__ ov,
                          int B) {
  const int TPB = (AN / 16) * (AN / 16);          // tiles per batch = 6400
  int wave = (blockIdx.x * blockDim.x + threadIdx.x) >> 5;
  int lane = threadIdx.x & 31;
  if (wave >= B * TPB) return;
  int b = wave / TPB;
  int t = wave - b * TPB;
  int ti = t / (AN / 16), tj = t - ti * (AN / 16);
  int i0 = ti * 16, j0 = tj * 16;
  int half = lane >> 4, n = lane & 15;

  // column (j) box for this lane
  const float* cb = boxes + (((size_t)b * AN) + j0 + n) * 4;
  float c0 = cb[0], c1 = cb[1], c2 = cb[2], c3 = cb[3];
  float areaI = area[(size_t)b * AN + i0 + n];
  float areaJ = area[(size_t)b * AN + j0 + n];

  // A frag (16x4 f32): lanes 0-15 carry K=0,1 ; lanes 16-31 carry K=2,3 (zero)
  v2f af, bf;
  af.x = (half == 0) ? areaI : 0.f;   // A[m][0] = area_i
  af.y = (half == 0) ? 1.0f  : 0.f;   // A[m][1] = 1
  bf.x = (half == 0) ? 1.0f  : 0.f;   // B[0][n] = 1
  bf.y = (half == 0) ? areaJ : 0.f;   // B[1][n] = area_j
  v8f acc = {};
  acc = __builtin_amdgcn_wmma_f32_16x16x4_f32(
      /*neg_a=*/false, af, /*neg_b=*/false, bf,
      /*c_mod=*/(short)0, acc, /*reuse_a=*/false, /*reuse_b=*/false);
  // acc[v] = area_i[i0+M] + area_j[j0+n],  M = v + 8*half

  uint32_t mv[8];
  #pragma unroll
  for (int v = 0; v < 8; ++v) {
    int M = v + 8 * half;
    const float* rb = boxes + (((size_t)b * AN) + i0 + M) * 4;
    float r0 = rb[0], r1 = rb[1], r2 = rb[2], r3 = rb[3];
    float lx = fmaxf(r0, c0), ly = fmaxf(r1, c1);
    float rx = fminf(r2, c2), ry = fminf(r3, c3);
    float wx = fmaxf(rx - lx, 0.f), wy = fmaxf(ry - ly, 0.f);
    float inter = wx * wy;
    float uni = acc[v] - inter;
    bool pred = (inter / uni) > 0.45f;
    mv[v] = __builtin_amdgcn_ballot_w32(pred);  // lo16: row i0+v, hi16: i0+v+8
  }

  // Route row halfwords: lanes 0-7 store rows i0+v (low halves),
  // lanes 16-23 store rows i0+8+v (high halves). One predicated store.
  int v = lane & 7;
  uint32_t x0 = (v & 1) ? mv[1] : mv[0];
  uint32_t x1 = (v & 1) ? mv[3] : mv[2];
  uint32_t x2 = (v & 1) ? mv[5] : mv[4];
  uint32_t x3 = (v & 1) ? mv[7] : mv[6];
  uint32_t y0 = (v & 2) ? x1 : x0;
  uint32_t y1 = (v & 2) ? x3 : x2;
  uint32_t z  = (v & 4) ? y1 : y0;
  uint16_t hw = (uint16_t)(half ? (z >> 16) : (z & 0xFFFFu));
  if ((lane & 15) < 8) {
    int row = i0 + v + 8 * half;
    uint16_t* p = (uint16_t*)(ov + ((size_t)b * AN + row) * ROWW);
    p[tj] = hw;
  }
}

// ---------------- K3: per-(b,c) descending argsort (bitonic, N=2048) --------
__global__ void k_sort(const float* __restrict__ cls,
                       int* __restrict__ order) {
  __shared__ float ks[2048];
  __shared__ int   vs[2048];
  int seq = blockIdx.x;                       // b*Cf + c
  const float* s = cls + (size_t)seq * AN;
  for (int i = threadIdx.x; i < 2048; i += blockDim.x) {
    ks[i] = (i < AN) ? s[i] : -3.0e38f;
    vs[i] = i;
  }
  __syncthreads();
  for (int k = 2; k <= 2048; k <<= 1) {
    for (int j = k >> 1; j > 0; j >>= 1) {
      for (int t = threadIdx.x; t < 1024; t += blockDim.x) {
        int i  = ((t & ~(j - 1)) << 1) | (t & (j - 1));
        int ix = i | j;
        bool dir = ((i & k) == 0);            // descending segment
        float ka = ks[i], kb = ks[ix];
        int   va = vs[i], vb = vs[ix];
        bool before = (ka > kb) || (ka == kb && va < vb); // (score desc, idx asc)
        if (before != dir) { ks[i] = kb; ks[ix] = ka; vs[i] = vb; vs[ix] = va; }
      }
      __syncthreads();
    }
  }
  for (int i = threadIdx.x; i < AN; i += blockDim.x)
    order[(size_t)seq * AN + i] = vs[i];
}

// ---------------- K4: sequential NMS scan, one wave per (b,c) ---------------
// Latency plan: valid bits precomputed in registers; order indices staged in
// registers per 32-step chunk; overlap rows prefetched one chunk ahead so the
// per-step dependency chain is ds_bpermute + ALU + a WGP-cache-hit row OR.
__global__ void k_nms(const float* __restrict__ cls,
                      const int* __restrict__ order,
                      const uint32_t* __restrict__ ov,
                      float* __restrict__ masked) {
  int seq = blockIdx.x;                       // b*Cf + c
  int b = seq / CF, c = seq - b * CF;
  int lane = threadIdx.x;                     // blockDim = 32
  const float* sc = cls + (size_t)seq * AN;
  const int*   od = order + (size_t)seq * AN;
  const uint32_t* ovb = ov + (size_t)b * AN * ROWW;

  // suppress + valid bitmasks: lane owns words 2*lane, 2*lane+1 (lane < 20)
  uint32_t s0 = 0, s1 = 0, vl0 = 0, vl1 = 0;
  if (lane < ROWW / 2) {
    #pragma unroll
    for (int t = 0; t < 2; ++t) {
      int w = 2 * lane + t;
      uint32_t bits = 0;
      for (int bb = 0; bb < 32; ++bb)
        bits |= (sc[w * 32 + bb] > 0.01f) ? (1u << bb) : 0u;
      if (t) vl1 = bits; else vl0 = bits;
    }
  }

  int cidx = od[lane];                        // chunk-0 indices in registers
  {                                            // prefetch chunk-0 rows (160 B)
    const char* rp = (const char*)(ovb + (size_t)cidx * ROWW);
    __builtin_prefetch(rp, 0, 1);
    __builtin_prefetch(rp + 128, 0, 1);
  }

  for (int r0 = 0; r0 < AN; r0 += 32) {
    int nidx = cidx;
    if (r0 + 32 < AN) {
      nidx = od[r0 + 32 + lane];
      const char* rp = (const char*)(ovb + (size_t)nidx * ROWW);
      __builtin_prefetch(rp, 0, 1);
      __builtin_prefetch(rp + 128, 0, 1);
    }
    #pragma unroll 4
    for (int s = 0; s < 32; ++s) {
      int idx  = __builtin_amdgcn_ds_bpermute(s << 2, cidx);  // broadcast
      int word = idx >> 5, bit = idx & 31;
      int owner = word >> 1;
      uint32_t sw = (word & 1) ? s1 : s0;
      uint32_t vw = (word & 1) ? vl1 : vl0;
      int cand = (int)(((~sw & vw) >> bit) & 1u);             // !sup && valid
      int bcast = __builtin_amdgcn_ds_bpermute(owner << 2, cand);
      if (bcast != 0) {                                       // wave-uniform
        if (lane < ROWW / 2) {
          const uint2* row = (const uint2*)(ovb + (size_t)idx * ROWW) + lane;
          uint2 rw = *row;
          s0 |= rw.x;
          s1 |= rw.y;
        }
        if (lane == owner) {
          if (word & 1) s1 &= ~(1u << bit); else s0 &= ~(1u << bit);
        }
      }
    }
    cidx = nidx;
  }

  // expand keep mask into masked scores (-1 = dropped)
  #pragma unroll
  for (int t2 = 0; t2 < 2; ++t2) {
    int w = 2 * lane + t2;
    if (w * 32 >= AN) break;
    uint32_t sw = t2 ? s1 : s0;
    uint32_t vw = t2 ? vl1 : vl0;
    for (int bit2 = 0; bit2 < 32; ++bit2) {
      int a = w * 32 + bit2;
      float sv = sc[a];
      bool keep = ((((~sw) & vw) >> bit2) & 1u) != 0u;
      masked[(size_t)b * (CF * AN) + (size_t)c * AN + a] = keep ? sv : -1.0f;
    }
  }
}

// ---------------- K5: per-batch top-K (bitonic over 32768, 256 KB LDS) ------
__global__ void k_topk(const float* __restrict__ masked,
                       const float* __restrict__ boxes,
                       float* __restrict__ out,
                       int B, int K) {
  extern __shared__ unsigned long long key[];   // 32768 entries = 256 KB
  const int M = CF * AN;                        // 25600
  const int N = 32768;
  int b = blockIdx.x;
  const float* m = masked + (size_t)b * M;
  for (int i = threadIdx.x; i < N; i += blockDim.x) {
    float v = (i < M) ? m[i] : -3.0e38f;
    uint32_t bits = __float_as_uint(v);
    uint32_t sortable = (bits & 0x80000000u) ? ~bits : (bits | 0x80000000u);
    key[i] = ((unsigned long long)sortable << 32) | (uint32_t)(~(uint32_t)i);
  }
  __syncthreads();
  for (int k = 2; k <= N; k <<= 1) {
    for (int j = k >> 1; j > 0; j >>= 1) {
      for (int t = threadIdx.x; t < N / 2; t += blockDim.x) {
        int i  = ((t & ~(j - 1)) << 1) | (t & (j - 1));
        int ix = i | j;
        bool dir = ((i & k) == 0);              // descending
        unsigned long long a = key[i], c2 = key[ix];
        bool before = a > c2;
        if (before != dir) { key[i] = c2; key[ix] = a; }
      }
      __syncthreads();
    }
  }
  // outputs concatenated flat: boxes [B,K,4] | labels [B,K] (int32) | scores [B,K]
  float* ob = out;
  int*   ol = (int*)(out + (size_t)B * K * 4);
  float* os = out + (size_t)B * K * 4 + (size_t)B * K;
  for (int kk = threadIdx.x; kk < K; kk += blockDim.x) {
    unsigned long long e = key[kk];
    uint32_t sortable = (uint32_t)(e >> 32);
    uint32_t bits = (sortable & 0x80000000u) ? (sortable ^ 0x80000000u) : ~sortable;
    float v = __uint_as_float(bits);
    int idx = (int)(~(uint32_t)e);
    bool dv = v > 0.0f;
    int lab = idx / AN + 1;
    int anc = idx - (lab - 1) * AN;
    const float* bx = boxes + ((size_t)b * AN + anc) * 4;
    float bb0 = dv ? bx[0] : 0.f, bb1 = dv ? bx[1] : 0.f;
    float bb2 = dv ? bx[2] : 0.f, bb3 = dv ? bx[3] : 0.f;
    size_t o = (size_t)b * K + kk;
    ob[o * 4 + 0] = bb0; ob[o * 4 + 1] = bb1;
    ob[o * 4 + 2] = bb2; ob[o * 4 + 3] = bb3;
    ol[o] = dv ? lab : 0;
    os[o] = dv ? v : 0.0f;
  }
}

// ---------------------------------------------------------------------------
extern "C" void kernel_launch(void* const* d_in, const int* in_sizes, int n_in,
                              void* d_out, int out_size, void* d_ws, size_t ws_size,
                              hipStream_t stream) {
  const float* locs    = (const float*)d_in[0];
  const float* scores  = (const float*)d_in[1];
  const float* anchors = (const float*)d_in[2];
  (void)n_in; (void)ws_size;

  int A = in_sizes[2] / 4;                    // 1280
  int B = in_sizes[0] / (A * 4);              // 16
  int K = out_size / (B * 6);                 // 200 (4 box + 1 label + 1 score)

  // workspace partition (all L2-resident, ~8.6 MB total)
  char* p = (char*)d_ws;
  auto take = [&](size_t bytes) { void* r = (void*)p; p += (bytes + 255) & ~(size_t)255; return r; };
  float*    boxes  = (float*)take((size_t)B * A * 4 * sizeof(float));
  float*    area   = (float*)take((size_t)B * A * sizeof(float));
  float*    cls    = (float*)take((size_t)B * CF * A * sizeof(float));
  int*      order  = (int*)take((size_t)B * CF * A * sizeof(int));
  uint32_t* ov     = (uint32_t*)take((size_t)B * A * ROWW * sizeof(uint32_t));
  float*    masked = (float*)take((size_t)B * CF * A * sizeof(float));

  int n1 = B * A;
  k_decode<<<(n1 + 255) / 256, 256, 0, stream>>>(locs, scores, anchors,
                                                 boxes, area, cls, B);
  int waves = B * (A / 16) * (A / 16);        // 102400 tiles, 8 waves/block
  k_overlap<<<(waves + 7) / 8, 256, 0, stream>>>(boxes, area, ov, B);
  k_sort<<<B * CF, 256, 0, stream>>>(cls, order);
  k_nms<<<B * CF, 32, 0, stream>>>(cls, order, ov, masked);
  k_topk<<<B, 1024, 262144, stream>>>(masked, boxes, (float*)d_out, B, K);
}